// NPair_25795573579970
// MI455X (gfx1250) — compile-verified
//
#include <hip/hip_runtime.h>

typedef __attribute__((ext_vector_type(2))) float v2f;
typedef __attribute__((ext_vector_type(8))) float v8f;

#define DIM 256
#define NROWS 8192                 // anchor rows (= positive rows)
#define TILE 16
#define KSTEPS (DIM / 4)           // 64 WMMA f32 16x16x4 steps per tile
#define WAVES_PER_BLOCK 4
#define BLOCK_THREADS (WAVES_PER_BLOCK * 32)
#define ROWS_PER_BLOCK (WAVES_PER_BLOCK * TILE)   // 64
#define LDS_PAD 260                // 1040 B row stride: 16B-aligned, bank stride 4
#define NTILES (NROWS / TILE)      // 512

// CDNA5 async DMA: global -> LDS, b128 chunks, tracked by ASYNCcnt.
// Stages positives rows n0..n0+15 (all 256 floats) into LDS buffer `buf`.
__device__ __forceinline__ void async_stage_tile(const float* __restrict__ net,
                                                 int n0, float* buf, int tid)
{
#pragma unroll
    for (int i = 0; i < (TILE * DIM / 4) / BLOCK_THREADS; ++i) {   // 8 chunks/thread
        int f  = tid + i * BLOCK_THREADS;      // b128 chunk index in tile
        int nl = f >> 6;                       // 64 x b128 per row
        int q  = f & 63;
        const float* g = net + (size_t)(2 * (n0 + nl) + 1) * DIM + 4 * q;
        unsigned lds = (unsigned)(unsigned long long)(buf + nl * LDS_PAD + 4 * q);
        asm volatile("global_load_async_to_lds_b128 %0, %1, off"
                     :: "v"(lds), "v"(g) : "memory");
    }
}

__device__ __forceinline__ void wait_async_all()
{
    asm volatile("s_wait_asynccnt 0x0" ::: "memory");
}

__global__ __launch_bounds__(BLOCK_THREADS)
void npair_lse_wmma_f32(const float* __restrict__ net, float* __restrict__ out)
{
    __shared__ alignas(16) float bt[2][TILE * LDS_PAD];   // double-buffered B tile

    const int tid   = threadIdx.x;
    const int lane  = tid & 31;
    const int wave  = tid >> 5;
    const int half  = lane >> 4;           // 0: lanes 0-15, 1: lanes 16-31
    const int lm    = lane & 15;
    const int kbase = 2 * half;            // fragment K offset for this half

    const int row0 = (blockIdx.x * WAVES_PER_BLOCK + wave) * TILE;

    // ---- A fragments: anchor rows row0..row0+15, resident in registers ----
    v2f a[KSTEPS];
    {
        const float* arow = net + (size_t)(2 * (row0 + lm)) * DIM + kbase;
#pragma unroll
        for (int kk = 0; kk < KSTEPS; ++kk)
            a[kk] = *(const v2f*)(arow + 4 * kk);
    }

    // online logsumexp state (per lane: 8 rows x column class n === lm mod 16)
    float m[8], s[8];
#pragma unroll
    for (int r = 0; r < 8; ++r) { m[r] = -__builtin_inff(); s[r] = 0.0f; }
    v8f diagc = {};
    const int diag_tile = row0 / TILE;

    // preload tile 0
    async_stage_tile(net, 0, &bt[0][0], tid);
    wait_async_all();
    __syncthreads();

    for (int t = 0; t < NTILES; ++t) {
        // overlap: DMA tile t+1 into the other buffer while computing tile t
        if (t + 1 < NTILES)
            async_stage_tile(net, (t + 1) * TILE, &bt[(t + 1) & 1][0], tid);

        // 16x16 score tile over K=256: 4 independent accumulator chains
        v8f c0 = {}, c1 = {}, c2 = {}, c3 = {};
        const float* bbase = &bt[t & 1][lm * LDS_PAD + kbase];
#pragma unroll
        for (int kk = 0; kk < KSTEPS; kk += 4) {
            v2f b0 = *(const v2f*)(bbase + 4 * (kk + 0));
            v2f b1 = *(const v2f*)(bbase + 4 * (kk + 1));
            v2f b2 = *(const v2f*)(bbase + 4 * (kk + 2));
            v2f b3 = *(const v2f*)(bbase + 4 * (kk + 3));
            c0 = __builtin_amdgcn_wmma_f32_16x16x4_f32(false, a[kk + 0], false, b0,
                                                       (short)0, c0, false, false);
            c1 = __builtin_amdgcn_wmma_f32_16x16x4_f32(false, a[kk + 1], false, b1,
                                                       (short)0, c1, false, false);
            c2 = __builtin_amdgcn_wmma_f32_16x16x4_f32(false, a[kk + 2], false, b2,
                                                       (short)0, c2, false, false);
            c3 = __builtin_amdgcn_wmma_f32_16x16x4_f32(false, a[kk + 3], false, b3,
                                                       (short)0, c3, false, false);
        }
        v8f c = (c0 + c1) + (c2 + c3);

        if (t == diag_tile) diagc = c;     // wave-uniform branch

#pragma unroll
        for (int r = 0; r < 8; ++r) {
            float v  = c[r];
            float nm = fmaxf(m[r], v);
            s[r] = s[r] * __expf(m[r] - nm) + __expf(v - nm);
            m[r] = nm;
        }

        // tile t+1 landed, everyone done reading tile t's buffer for next round
        wait_async_all();
        __syncthreads();
    }

    // merge LSE across the 16 lanes of each half (xor 8,4,2,1 stays within a half)
#pragma unroll
    for (int r = 0; r < 8; ++r) {
#pragma unroll
        for (int off = 8; off >= 1; off >>= 1) {
            float om = __shfl_xor(m[r], off, 32);
            float os = __shfl_xor(s[r], off, 32);
            float nm = fmaxf(m[r], om);
            s[r] = s[r] * __expf(m[r] - nm) + os * __expf(om - nm);
            m[r] = nm;
        }
    }

    // loss_i = LSE_i - s_ii ; diagonal: row r (half0) at lane r, row 8+r (half1)
    // at lane 24+r, both in vgpr slot r of the saved diagonal tile
    float loss = 0.0f;
#pragma unroll
    for (int r = 0; r < 8; ++r) {
        float d0  = __shfl(diagc[r], r, 32);
        float d1  = __shfl(diagc[r], 24 + r, 32);
        float dia = half ? d1 : d0;
        loss += (m[r] + __logf(s[r])) - dia;
    }

    if (lane == 0 || lane == 16)
        atomicAdd(out, loss * (1.0f / (float)NROWS));
}

extern "C" void kernel_launch(void* const* d_in, const int* in_sizes, int n_in,
                              void* d_out, int out_size, void* d_ws, size_t ws_size,
                              hipStream_t stream)
{
    const float* net = (const float*)d_in[0];
    float* out = (float*)d_out;

    // d_out is poisoned by the harness; zero it every call (capture-legal).
    hipMemsetAsync(out, 0, sizeof(float), stream);

    dim3 grid(NROWS / ROWS_PER_BLOCK);   // 128 workgroups x 4 waves x 16 rows
    npair_lse_wmma_f32<<<grid, BLOCK_THREADS, 0, stream>>>(net, out);
}